// MASS_18897856102446
// MI455X (gfx1250) — compile-verified
//
#include <hip/hip_runtime.h>

typedef __attribute__((ext_vector_type(2))) float v2f;
typedef __attribute__((ext_vector_type(4))) float v4f;
typedef __attribute__((ext_vector_type(8))) float v8f;

#define B_DIM 1024
#define N_DIM 200
#define E_DIM 128
#define NTILE 13          // ceil(200/16)
#define NPAD  (NTILE * 16)
#define NPAIR (E_DIM / 8) // 16 chunk-pairs, 8 e-values each

#define WMMA_F32(A, Bf, C) \
    __builtin_amdgcn_wmma_f32_16x16x4_f32(false, (A), false, (Bf), (short)0, (C), false, false)

__global__ __launch_bounds__(256) void mass_wmma_kernel(
    const float* __restrict__ q,      // [B,E]
    const float* __restrict__ q_p,    // [B,E]
    const float* __restrict__ m,      // [B,N,E]
    const float* __restrict__ m_c,    // [B,N,E]
    const float* __restrict__ A1,     // [E,1]
    const float* __restrict__ A2,     // [E,1]
    const float* __restrict__ biases, // [B,N]
    const float* __restrict__ mask,   // [B,N]
    float* __restrict__ out)          // [B,N]
{
    __shared__ float Ul[E_DIM];
    __shared__ float Wl[E_DIM];
    __shared__ v4f   Bfr[NPAIR * 2 * 32];  // per-lane B fragments: {bu.x,bu.y,bw.x,bw.y}
    __shared__ float red[256];
    __shared__ float s_att[NPAD];          // -(att * mask)
    __shared__ float s_out[NPAD];          // out_dist (pre-softmax-weight)
    __shared__ float s_cst;

    const int b   = blockIdx.x;
    const int tid = threadIdx.x;

    // ---------- Phase 1a: per-batch U, W vectors and constant C ----------
    float cpart = 0.0f;
    if (tid < E_DIM) {
        float a1  = A1[tid];
        float a2  = A2[tid];
        float a1s = a1 * a1;
        float a2s = a2 * a2;
        float qe  = q  [(size_t)b * E_DIM + tid];
        float qpe = q_p[(size_t)b * E_DIM + tid];
        Wl[tid] = a1s + a2s;
        Ul[tid] = -2.0f * (a1s * qe + a2s * qpe);
        cpart   = a1s * qe * qe + a2s * qpe * qpe;
    }
    red[tid] = cpart;
    __syncthreads();
    for (int s = 128; s > 0; s >>= 1) {
        if (tid < s) red[tid] += red[tid + s];
        __syncthreads();
    }
    if (tid == 0) s_cst = red[0];
    __syncthreads();
    const float Cst = s_cst;

    // ---------- Phase 1b: bake masked B fragments in WMMA lane layout ----------
    // For chunk-pair p, sub-chunk j (two WMMA K-steps per pair), lane L:
    //   B column 0 carried by lanes with L%16==0; half = L>>4 selects K slots.
    //   e-values for (p, half, j): e = 8p + 4*half + 2j (+0, +1)
    for (int s = tid; s < NPAIR * 32; s += 256) {
        const int  p  = s >> 5;
        const int  L  = s & 31;
        const int  h  = L >> 4;
        const bool c0 = (L & 15) == 0;
        #pragma unroll
        for (int j = 0; j < 2; ++j) {
            const int e = p * 8 + 4 * h + 2 * j;
            v4f f;
            f.x = c0 ? Ul[e]     : 0.0f;
            f.y = c0 ? Ul[e + 1] : 0.0f;
            f.z = c0 ? Wl[e]     : 0.0f;
            f.w = c0 ? Wl[e + 1] : 0.0f;
            Bfr[(p * 2 + j) * 32 + L] = f;
        }
    }
    __syncthreads();

    // ---------- Phase 2: 16-row tiles via V_WMMA_F32_16X16X4_F32 ----------
    const int  wave = tid >> 5;
    const int  lane = tid & 31;
    const int  mrow = lane & 15;   // output row within tile (A-matrix M)
    const int  half = lane >> 4;   // K-slot selector
    const bool col0 = (mrow == 0);

    for (int t = wave; t < NTILE; t += 8) {
        const int n0 = t * 16;
        int nr = n0 + mrow;
        if (nr > N_DIM - 1) nr = N_DIM - 1;   // clamp tail tile (masked on store)
        const float* mr  = m   + ((size_t)b * N_DIM + nr) * E_DIM;
        const float* mcr = m_c + ((size_t)b * N_DIM + nr) * E_DIM;

        v8f accA = {};   // sum_e  m·U + m^2·W     (att path)
        v8f accC = {};   // same with m_c           (out path)
        #pragma unroll
        for (int p = 0; p < NPAIR; ++p) {
            const int eb = p * 8 + 4 * half;
            v4f am  = *(const v4f*)(mr  + eb);   // global_load_b128
            v4f acm = *(const v4f*)(mcr + eb);   // global_load_b128
            const v4f f0 = Bfr[(p * 2 + 0) * 32 + lane];  // ds_load_b128
            const v4f f1 = Bfr[(p * 2 + 1) * 32 + lane];  // ds_load_b128

            v2f a0  = {am.x,  am.y},  a1v = {am.z,  am.w};
            v2f ac0 = {acm.x, acm.y}, ac1 = {acm.z, acm.w};
            v2f bu0 = {f0.x, f0.y},   bw0 = {f0.z, f0.w};
            v2f bu1 = {f1.x, f1.y},   bw1 = {f1.z, f1.w};
            v2f a0s  = a0  * a0,  a1s  = a1v * a1v;
            v2f ac0s = ac0 * ac0, ac1s = ac1 * ac1;

            accA = WMMA_F32(a0,   bu0, accA);
            accA = WMMA_F32(a0s,  bw0, accA);
            accA = WMMA_F32(a1v,  bu1, accA);
            accA = WMMA_F32(a1s,  bw1, accA);
            accC = WMMA_F32(ac0,  bu0, accC);
            accC = WMMA_F32(ac0s, bw0, accC);
            accC = WMMA_F32(ac1,  bu1, accC);
            accC = WMMA_F32(ac1s, bw1, accC);
        }
        // Extract D column 0: lane 0 -> rows 0..7, lane 16 -> rows 8..15
        if (col0) {
            const float* bb = biases + (size_t)b * N_DIM;
            const float* mk = mask   + (size_t)b * N_DIM;
            #pragma unroll
            for (int v = 0; v < 8; ++v) {
                const int idx = half * 8 + v;
                const int n   = n0 + idx;
                float b2 = 0.0f, mv = 0.0f;
                if (n < N_DIM) { b2 = 2.0f * bb[n]; mv = mk[n]; }
                const float att = Cst + b2 + accA[v];
                const float od  = (Cst + b2 + accC[v]) * mv;
                s_att[n0 + idx] = -(att * mv);
                s_out[n0 + idx] = od;
            }
        }
    }
    __syncthreads();

    // ---------- Phase 3: softmax over n, final weighting ----------
    float sval = -__builtin_inff();
    if (tid < N_DIM) sval = s_att[tid];
    red[tid] = sval;
    __syncthreads();
    for (int s = 128; s > 0; s >>= 1) {
        if (tid < s) red[tid] = fmaxf(red[tid], red[tid + s]);
        __syncthreads();
    }
    const float mx = red[0];
    __syncthreads();
    float e = 0.0f;
    if (tid < N_DIM) e = expf(sval - mx);
    red[tid] = e;
    __syncthreads();
    for (int s = 128; s > 0; s >>= 1) {
        if (tid < s) red[tid] += red[tid + s];
        __syncthreads();
    }
    const float inv = 1.0f / red[0];
    if (tid < N_DIM) out[(size_t)b * N_DIM + tid] = s_out[tid] * e * inv;
}

extern "C" void kernel_launch(void* const* d_in, const int* in_sizes, int n_in,
                              void* d_out, int out_size, void* d_ws, size_t ws_size,
                              hipStream_t stream) {
    (void)in_sizes; (void)n_in; (void)d_ws; (void)ws_size; (void)out_size;
    const float* q   = (const float*)d_in[0];
    const float* q_p = (const float*)d_in[1];
    const float* mm  = (const float*)d_in[2];
    const float* mc  = (const float*)d_in[3];
    const float* A1  = (const float*)d_in[4];
    const float* A2  = (const float*)d_in[5];
    const float* bi  = (const float*)d_in[6];
    const float* ms  = (const float*)d_in[7];
    float* outp = (float*)d_out;
    mass_wmma_kernel<<<B_DIM, 256, 0, stream>>>(q, q_p, mm, mc, A1, A2, bi, ms, outp);
}